// NERBertCRF_67946382623036
// MI455X (gfx1250) — compile-verified
//
#include <hip/hip_runtime.h>
#include <hip/hip_bf16.h>

// ---------------- problem constants ----------------
#define DD   768          // hidden dim
#define CC   37           // num classes (+PAD)
#define CP   40           // padded emission row stride (floats, 160B-aligned rows)
#define NP   48           // padded N for 3 x 16 WMMA tiles
#define BB   128          // batch
#define SS   512          // sequence
#define PAD_ID (CC - 1)

typedef float v2f __attribute__((ext_vector_type(2)));
typedef float v8f __attribute__((ext_vector_type(8)));

// =====================================================================
// Kernel 1: emissions = x @ W + b  via V_WMMA_F32_16X16X4_F32
//   grid: 512 blocks x 128 threads (4 waves); each wave owns TWO 16-row
//   M-tiles (B fragments amortized over both), all 3 N-tiles (N 37->48).
//   W staged in LDS as K-pairs: Wp[k/2][col][2] so each B fragment is a
//   single aligned ds_load_b64 (conflict-free across all 64 banks).
//   Bias folded into the WMMA C operand (accumulator init).
// =====================================================================
__global__ void emis_gemm(const float* __restrict__ x,
                          const float* __restrict__ W,
                          const float* __restrict__ bias,
                          float* __restrict__ em) {
  extern __shared__ float Wp[];               // [DD/2][NP][2] zero-padded W
  const int tid = threadIdx.x;                // 0..127

  // cooperative zero-padded, K-paired load of W into LDS
  for (int idx = tid; idx < DD * NP; idx += 128) {
    const int r   = idx & 1;                  // K parity within pair
    const int t   = idx >> 1;
    const int col = t % NP;
    const int k   = (t / NP) * 2 + r;
    Wp[idx] = (col < CC) ? W[k * CC + col] : 0.0f;
  }
  __syncthreads();

  const int wave = tid >> 5;                  // 0..3
  const int lane = tid & 31;
  const int half = lane >> 4;                 // 0: K0/K1 slice, 1: K2/K3 slice
  const int l16  = lane & 15;
  const long m0  = ((long)blockIdx.x * 8 + wave * 2) * 16;  // first of two tiles

  // bias per N-tile column (C-operand init -> D = sum(A*B) + bias)
  v8f acc[2][3];
#pragma unroll
  for (int n = 0; n < 3; ++n) {
    const int col = n * 16 + l16;
    const float bv = (col < CC) ? bias[col] : 0.0f;
#pragma unroll
    for (int r = 0; r < 8; ++r) { acc[0][n][r] = bv; acc[1][n][r] = bv; }
  }

  // A: lane holds row m0(+16)+l16; VGPR0/1 = K(+0,+1) half 0, K(+2,+3) half 1
  const float* __restrict__ xrow0 = x + (m0 + l16) * DD + 2 * half;
  const float* __restrict__ xrow1 = xrow0 + 16 * DD;

  for (int k = 0; k < DD; k += 4) {
    const v2f a0 = *(const v2f*)(xrow0 + k);  // 8B, 8B-aligned (k even)
    const v2f a1 = *(const v2f*)(xrow1 + k);
    // B fragments: K-pair (k+2*half, k+2*half+1) at column col, contiguous
    const float* __restrict__ bbase = Wp + ((k >> 1) + half) * (NP * 2);
#pragma unroll
    for (int n = 0; n < 3; ++n) {
      const v2f bf = *(const v2f*)(bbase + (n * 16 + l16) * 2);
      acc[0][n] = __builtin_amdgcn_wmma_f32_16x16x4_f32(
          false, a0, false, bf, (short)0, acc[0][n], false, false);
      acc[1][n] = __builtin_amdgcn_wmma_f32_16x16x4_f32(
          false, a1, false, bf, (short)0, acc[1][n], false, false);
    }
  }

  // D layout: VGPR r -> M = m0 + 16*tile + r + 8*half, N = n*16 + l16
#pragma unroll
  for (int tile = 0; tile < 2; ++tile) {
#pragma unroll
    for (int n = 0; n < 3; ++n) {
      const int col = n * 16 + l16;
      if (col < CP) {
#pragma unroll
        for (int r = 0; r < 8; ++r) {
          const long row = m0 + 16 * tile + r + 8 * half;
          em[row * CP + col] = acc[tile][n][r];
        }
      }
    }
  }
}

// =====================================================================
// Kernel 2: CRF per batch element (forward logsumexp + gold score +
// Viterbi + backtrack).  128 blocks x 64 threads (2 waves).
// Threads 0..36: one CRF state each. Thread 37: gold-path score.
// alpha/v double-buffered in LDS; backpointers kept entirely in LDS.
// =====================================================================
__global__ void __launch_bounds__(64)
crf_kernel(const float* __restrict__ em,
           const int*   __restrict__ labels,
           const int*   __restrict__ mask,
           const float* __restrict__ start,
           const float* __restrict__ endv,
           const float* __restrict__ trans,
           float* __restrict__ partial,      // [128]  score - logZ
           float* __restrict__ tags_out) {   // [128*512] as float
  __shared__ float trS[CC * CC];             // 5476 B
  __shared__ float alphaBuf[2][CP];
  __shared__ float vBuf[2][CP];
  __shared__ unsigned char bp[SS][CP];       // 20480 B backpointers
  __shared__ unsigned char tagseq[SS];
  __shared__ float sh_score;
  __shared__ float sh_logZ;
  __shared__ int   sh_last;

  const int b  = blockIdx.x;
  const int tx = threadIdx.x;                // 0..63

  for (int i = tx; i < CC * CC; i += 64) trS[i] = trans[i];

  const float* __restrict__ emb = em + (long)b * SS * CP;
  const int*   __restrict__ tg  = labels + b * SS;
  const int*   __restrict__ mk  = mask   + b * SS;

  // t = 0 (mask[.,0] always true)
  if (tx < CC) {
    const float v0 = start[tx] + emb[tx];
    alphaBuf[0][tx] = v0;
    vBuf[0][tx]     = v0;
  }
  int   prev_tag = 0;
  float score    = 0.0f;
  if (tx == 37) {
    prev_tag = tg[0];
    score    = start[prev_tag] + emb[prev_tag];
  }
  __syncthreads();

  int cur = 0;
  for (int t = 1; t < SS; ++t) {
    const int nxt = cur ^ 1;
    const int mb  = mk[t];
    const float* __restrict__ emt = emb + t * CP;
    if (tx < CC) {
      const float* al = alphaBuf[cur];
      const float* vv = vBuf[cur];
      float maxa = -3.4e38f, maxv = -3.4e38f;
      int   arg  = 0;
      // fused max-pass: logsumexp max + Viterbi max/argmax (first-max wins)
      for (int p = 0; p < CC; ++p) {
        const float T  = trS[p * CC + tx];
        const float ca = al[p] + T;
        maxa = fmaxf(maxa, ca);
        const float cv = vv[p] + T;
        if (cv > maxv) { maxv = cv; arg = p; }
      }
      float s = 0.0f;
      for (int p = 0; p < CC; ++p)
        s += __expf(al[p] + trS[p * CC + tx] - maxa);
      const float e  = emt[tx];
      const float na = maxa + __logf(s) + e;
      const float nv = maxv + e;
      alphaBuf[nxt][tx] = mb ? na : al[tx];
      vBuf[nxt][tx]     = mb ? nv : vv[tx];
      bp[t][tx] = (unsigned char)(mb ? arg : tx);   // identity at padded steps
    } else if (tx == 37) {
      if (mb) {
        const int tt = tg[t];
        score   += trS[prev_tag * CC + tt] + emt[tt];
        prev_tag = tt;
      }
    }
    __syncthreads();
    cur = nxt;
  }

  if (tx == 37) sh_score = score + endv[prev_tag];

  if (tx == 0) {
    const float* al = alphaBuf[cur];
    float m = -3.4e38f;
    for (int c = 0; c < CC; ++c) m = fmaxf(m, al[c] + endv[c]);
    float s = 0.0f;
    for (int c = 0; c < CC; ++c) s += __expf(al[c] + endv[c] - m);
    sh_logZ = m + __logf(s);

    const float* vv = vBuf[cur];
    float bm = -3.4e38f; int bi = 0;
    for (int c = 0; c < CC; ++c) {
      const float cv = vv[c] + endv[c];
      if (cv > bm) { bm = cv; bi = c; }       // first max, matches jnp.argmax
    }
    sh_last = bi;
  }
  __syncthreads();

  if (tx == 0) {
    partial[b] = sh_score - sh_logZ;
    int tag = sh_last;
    tagseq[SS - 1] = (unsigned char)tag;
    for (int t = SS - 1; t >= 1; --t) {
      tag = bp[t][tag];
      tagseq[t - 1] = (unsigned char)tag;
    }
  }
  __syncthreads();

  for (int t = tx; t < SS; t += 64) {
    const int tv = mk[t] ? (int)tagseq[t] : PAD_ID;
    tags_out[(long)b * SS + t] = (float)tv;
  }
}

// =====================================================================
// Kernel 3: deterministic serial sum of per-batch partials -> out[0]
// =====================================================================
__global__ void ll_reduce(const float* __restrict__ partial,
                          float* __restrict__ out) {
  if (threadIdx.x == 0) {
    float s = 0.0f;
    for (int i = 0; i < BB; ++i) s += partial[i];
    out[0] = s;
  }
}

// =====================================================================
extern "C" void kernel_launch(void* const* d_in, const int* in_sizes, int n_in,
                              void* d_out, int out_size, void* d_ws, size_t ws_size,
                              hipStream_t stream) {
  const float* x     = (const float*)d_in[0];   // [128,512,768]
  const float* W     = (const float*)d_in[1];   // [768,37]
  const float* bias  = (const float*)d_in[2];   // [37]
  const float* start = (const float*)d_in[3];   // [37]
  const float* endv  = (const float*)d_in[4];   // [37]
  const float* trans = (const float*)d_in[5];   // [37,37]
  const int*   labels= (const int*)d_in[6];     // [128,512]
  const int*   mask  = (const int*)d_in[7];     // [128,512]
  float* out = (float*)d_out;                   // [1 + 128*512]

  float* em      = (float*)d_ws;                                  // [65536,40]
  float* partial = (float*)((char*)d_ws + (size_t)BB * SS * CP * sizeof(float));

  const size_t lds_bytes = (size_t)DD * NP * sizeof(float);       // 147456 B
  emis_gemm<<<dim3(512), dim3(128), lds_bytes, stream>>>(x, W, bias, em);
  crf_kernel<<<dim3(BB), dim3(64), 0, stream>>>(em, labels, mask, start, endv,
                                                trans, partial, out + 1);
  ll_reduce<<<dim3(1), dim3(32), 0, stream>>>(partial, out);
}